// DCGN_78967268704510
// MI455X (gfx1250) — compile-verified
//
#include <hip/hip_runtime.h>
#include <hip/hip_bf16.h>

// ---------------------------------------------------------------------------
// DCGN forward for MI455X (gfx1250, wave32).
// adj is provably diagonal (faithful to the reference's buggy loop), so the
// "graph propagation" collapses to a per-row scale; the heavy work is two
// fp32 GEMMs done with V_WMMA_F32_16X16X4_F32.
// ---------------------------------------------------------------------------

typedef __attribute__((ext_vector_type(2))) float v2f;
typedef __attribute__((ext_vector_type(8))) float v8f;

#define SLOPE 0.01f
#define EPSN  1e-8f

// ===========================================================================
// Stage prep: per window of P=4 rows of length F:
//   logits_p = dot(x_p, poolw) + poolb        (4 dots, tree-reduced)
//   att      = softmax(logits)                (4-way)
//   pooled_f = sum_p att_p * x_p[f]           (only its squared norm is kept)
//   conv_f   = sum_p convw[p,f] * x_p[f]
//   scale    = dinv*diag*dinv with diag = cos(pooled,pooled)*[2s<S] + 1
//   g_f      = scale * conv_f
// One 256-thread block per window; the 4 rows live in LDS (read x once).
// ===========================================================================
template<int F, int S, int FPT>
__global__ void __launch_bounds__(256)
stage_prep_kernel(const float* __restrict__ x,      // [numWindows*4, F]
                  const float* __restrict__ convw,  // [4, F]
                  const float* __restrict__ poolw,  // [F]
                  const float* __restrict__ poolb,  // [1]
                  float* __restrict__ g)            // [numWindows, F]
{
    __shared__ float lx[4 * F];
    __shared__ float red[256];
    __shared__ float att_s[4];
    __shared__ float scale_s;

    const int tid = threadIdx.x;
    const int w   = blockIdx.x;          // window index (b*S + s)
    const int s   = w % S;
    const float* __restrict__ xr = x + (size_t)w * 4 * F;

    // ---- pass 1: stage rows into LDS, partial pool-logit dots -------------
    float d0 = 0.f, d1 = 0.f, d2 = 0.f, d3 = 0.f;
    for (int f = tid; f < F; f += 256) {
        const float pwv = poolw[f];
        float v0 = xr[0 * F + f];
        float v1 = xr[1 * F + f];
        float v2 = xr[2 * F + f];
        float v3 = xr[3 * F + f];
        lx[0 * F + f] = v0;
        lx[1 * F + f] = v1;
        lx[2 * F + f] = v2;
        lx[3 * F + f] = v3;
        d0 += v0 * pwv; d1 += v1 * pwv; d2 += v2 * pwv; d3 += v3 * pwv;
    }
    float dp[4] = {d0, d1, d2, d3};
    #pragma unroll
    for (int p = 0; p < 4; ++p) {
        __syncthreads();
        red[tid] = dp[p];
        __syncthreads();
        for (int off = 128; off > 0; off >>= 1) {
            if (tid < off) red[tid] += red[tid + off];
            __syncthreads();
        }
        if (tid == 0) att_s[p] = red[0];
    }
    __syncthreads();

    // ---- softmax over the 4 logits (thread 0) -----------------------------
    if (tid == 0) {
        const float b = poolb[0];
        float l0 = att_s[0] + b, l1 = att_s[1] + b;
        float l2 = att_s[2] + b, l3 = att_s[3] + b;
        float m  = fmaxf(fmaxf(l0, l1), fmaxf(l2, l3));
        float e0 = expf(l0 - m), e1 = expf(l1 - m);
        float e2 = expf(l2 - m), e3 = expf(l3 - m);
        float inv = 1.0f / (e0 + e1 + e2 + e3);
        att_s[0] = e0 * inv; att_s[1] = e1 * inv;
        att_s[2] = e2 * inv; att_s[3] = e3 * inv;
    }
    __syncthreads();
    const float a0 = att_s[0], a1 = att_s[1], a2 = att_s[2], a3 = att_s[3];

    // ---- pass 2: pooled squared-norm + per-feature conv -------------------
    float conv_r[FPT];
    float nsq = 0.f;
    #pragma unroll
    for (int i = 0; i < FPT; ++i) {
        const int f = tid + i * 256;
        float cv = 0.f;
        if (f < F) {
            float v0 = lx[0 * F + f];
            float v1 = lx[1 * F + f];
            float v2 = lx[2 * F + f];
            float v3 = lx[3 * F + f];
            float pooled = a0 * v0 + a1 * v1 + a2 * v2 + a3 * v3;
            nsq += pooled * pooled;
            cv = convw[0 * F + f] * v0 + convw[1 * F + f] * v1 +
                 convw[2 * F + f] * v2 + convw[3 * F + f] * v3;
        }
        conv_r[i] = cv;
    }
    __syncthreads();
    red[tid] = nsq;
    __syncthreads();
    for (int off = 128; off > 0; off >>= 1) {
        if (tid < off) red[tid] += red[tid + off];
        __syncthreads();
    }
    if (tid == 0) {
        const float t    = red[0];
        const float nrm  = fmaxf(sqrtf(t), EPSN);
        const float cs   = t / (nrm * nrm);
        const float diag = ((2 * s) < S ? cs : 0.0f) + 1.0f;
        const float dinv = 1.0f / sqrtf(diag);
        scale_s = dinv * diag * dinv;
    }
    __syncthreads();
    const float sc = scale_s;

    float* __restrict__ go = g + (size_t)w * F;
    #pragma unroll
    for (int i = 0; i < FPT; ++i) {
        const int f = tid + i * 256;
        if (f < F) go[f] = sc * conv_r[i];
    }
}

// ===========================================================================
// fp32 WMMA GEMM: C[M,N] = leaky_relu(A[M,K] @ B[K,N] + bias[N])
// Block tile 128x64, 8 waves; each wave owns a 16x64 strip (four 16x16 wmma
// accumulators, A fragment reused 4x). K staged through LDS in 32-wide slabs.
// A slab: row-major, stride 34 (K-pairs adjacent -> ds_load_2addr_b64).
// B slab: k-pair interleaved float2 so each fragment {B[k][n],B[k+1][n]} is
// one aligned ds_load_b64 (no register repacking).
// M must be a multiple of 128; N and K may have tails (zero-padded).
// ===========================================================================
#define LDA  34   // floats per A row (32 k + 2 pad)
#define LDB2 65   // float2 per B k-pair row (64 cols + 1 pad)

__global__ void __launch_bounds__(256)
gemm_bias_lrelu_kernel(const float* __restrict__ A, const float* __restrict__ Bm,
                       const float* __restrict__ bias, float* __restrict__ C,
                       int M, int N, int K)
{
    __shared__ float  at[128 * LDA];
    __shared__ float2 bt[16 * LDB2];

    const int tid  = threadIdx.x;
    const int m0   = blockIdx.y * 128;
    const int n0   = blockIdx.x * 64;
    const int lane = tid & 31;
    const int wid  = tid >> 5;
    const int wm   = wid * 16;         // wave's 16-row sub-tile
    const int lo   = lane & 15;
    const int hi   = lane >> 4;

    v8f acc[4] = {{}, {}, {}, {}};

    for (int k0 = 0; k0 < K; k0 += 32) {
        // ---- stage A slab: 128 rows x 32 k, float4 loads ------------------
        for (int i = tid; i < 128 * 8; i += 256) {
            const int r  = i >> 3;
            const int c4 = (i & 7) * 4;
            const int gk = k0 + c4;
            const float* src = A + (size_t)(m0 + r) * K + gk;
            float4 v;
            if (gk + 4 <= K) {
                v = *(const float4*)src;
            } else {
                v.x = (gk + 0 < K) ? src[0] : 0.f;
                v.y = (gk + 1 < K) ? src[1] : 0.f;
                v.z = (gk + 2 < K) ? src[2] : 0.f;
                v.w = 0.f;
            }
            float* dst = &at[r * LDA + c4];
            dst[0] = v.x; dst[1] = v.y; dst[2] = v.z; dst[3] = v.w;
        }
        // ---- stage B slab: 32 k x 64 cols, k-pair interleaved -------------
        for (int i = tid; i < 32 * 16; i += 256) {
            const int r  = i >> 4;           // k row within slab
            const int c4 = (i & 15) * 4;     // first column
            const int gk = k0 + r;
            const int gn = n0 + c4;
            float4 v = {0.f, 0.f, 0.f, 0.f};
            if (gk < K) {
                const float* src = Bm + (size_t)gk * N + gn;
                if (gn + 4 <= N) {
                    v = *(const float4*)src;
                } else {
                    if (gn + 0 < N) v.x = src[0];
                    if (gn + 1 < N) v.y = src[1];
                    if (gn + 2 < N) v.z = src[2];
                    if (gn + 3 < N) v.w = src[3];
                }
            }
            // element (r, c) lands at bt[(r/2)*LDB2 + c].{x if r even, y if r odd}
            float* dstf = ((float*)&bt[(r >> 1) * LDB2 + c4]) + (r & 1);
            dstf[0] = v.x; dstf[2] = v.y; dstf[4] = v.z; dstf[6] = v.w;
        }
        // speculative prefetch of the next K slab (global_prefetch_b8)
        if (k0 + 32 < K) {
            __builtin_prefetch(A + (size_t)(m0 + (tid & 127)) * K + k0 + 32, 0, 0);
            __builtin_prefetch(Bm + (size_t)(k0 + 32 + (tid & 31)) * N + n0, 0, 0);
        }
        __syncthreads();

        // ---- 8 k-steps x 4 tiles of V_WMMA_F32_16X16X4_F32 ----------------
        #pragma unroll
        for (int ks = 0; ks < 32; ks += 4) {
            const int ak = ks + 2 * hi;   // lanes 0-15: K=ks,ks+1 ; 16-31: ks+2,ks+3
            v2f a;
            a.x = at[(wm + lo) * LDA + ak + 0];
            a.y = at[(wm + lo) * LDA + ak + 1];
            #pragma unroll
            for (int t = 0; t < 4; ++t) {
                float2 bb = bt[((ks >> 1) + hi) * LDB2 + t * 16 + lo];
                v2f bf;
                bf.x = bb.x;
                bf.y = bb.y;
                acc[t] = __builtin_amdgcn_wmma_f32_16x16x4_f32(
                             false, a, false, bf, (short)0, acc[t], false, false);
            }
        }
        __syncthreads();
    }

    // ---- epilogue: bias + LeakyReLU, N-tail guarded -----------------------
    #pragma unroll
    for (int t = 0; t < 4; ++t) {
        const int col = n0 + t * 16 + lo;
        if (col < N) {
            const float bv = bias[col];
            #pragma unroll
            for (int r = 0; r < 8; ++r) {
                const int row = m0 + wm + r + 8 * hi;  // lanes>=16 -> M=r+8
                float v = acc[t][r] + bv;
                C[(size_t)row * N + col] = (v >= 0.f) ? v : SLOPE * v;
            }
        }
    }
}

// ===========================================================================
// Launcher
// ===========================================================================
extern "C" void kernel_launch(void* const* d_in, const int* in_sizes, int n_in,
                              void* d_out, int out_size, void* d_ws, size_t ws_size,
                              hipStream_t stream) {
    (void)in_sizes; (void)n_in; (void)out_size; (void)ws_size;

    const float* x       = (const float*)d_in[0];
    const float* conv1_w = (const float*)d_in[1];
    const float* pool1_w = (const float*)d_in[2];
    const float* pool1_b = (const float*)d_in[3];
    const float* prop1_W = (const float*)d_in[4];
    const float* prop1_B = (const float*)d_in[5];
    const float* conv2_w = (const float*)d_in[6];
    const float* pool2_w = (const float*)d_in[7];
    const float* pool2_b = (const float*)d_in[8];
    const float* prop2_W = (const float*)d_in[9];
    const float* prop2_B = (const float*)d_in[10];
    // d_in[11] = pooling_size (always 4 for these shapes)

    constexpr int B = 64, N = 512, F = 2048, HID = 1100, NCLASS = 512;
    constexpr int S1 = 128;          // N / 4
    constexpr int S2 = 32;           // S1 / 4
    constexpr int M1 = B * S1;       // 8192 GEMM1 rows
    constexpr int M2 = B * S2;       // 2048 GEMM2 rows

    // workspace partition (256B-aligned)
    char*  ws = (char*)d_ws;
    size_t o0 = 0;
    size_t o1 = o0 + ((size_t)M1 * F * sizeof(float) + 255) / 256 * 256;
    size_t o2 = o1 + ((size_t)M1 * HID * sizeof(float) + 255) / 256 * 256;
    float* g1 = (float*)(ws + o0);   // scaled conv1 rows   [M1, F]
    float* h1 = (float*)(ws + o1);   // stage-1 activations [M1, HID]
    float* g2 = (float*)(ws + o2);   // scaled conv2 rows   [M2, HID]
    float* out = (float*)d_out;      // [M2, NCLASS]

    // stage 1 prep: windows over x
    stage_prep_kernel<F, S1, 8><<<M1, 256, 0, stream>>>(
        x, conv1_w, pool1_w, pool1_b, g1);

    // h1 = leaky_relu(g1 @ prop1_W + prop1_B)   [8192,2048]x[2048,1100]
    {
        dim3 grid((HID + 63) / 64, M1 / 128);
        gemm_bias_lrelu_kernel<<<grid, 256, 0, stream>>>(
            g1, prop1_W, prop1_B, h1, M1, HID, F);
    }

    // stage 2 prep: windows over h1
    stage_prep_kernel<HID, S2, 5><<<M2, 256, 0, stream>>>(
        h1, conv2_w, pool2_w, pool2_b, g2);

    // out = leaky_relu(g2 @ prop2_W + prop2_B)  [2048,1100]x[1100,512]
    {
        dim3 grid(NCLASS / 64, M2 / 128);
        gemm_bias_lrelu_kernel<<<grid, 256, 0, stream>>>(
            g2, prop2_W, prop2_B, out, M2, NCLASS, HID);
    }
}